// GAT_51058571215471
// MI455X (gfx1250) — compile-verified
//
#include <hip/hip_runtime.h>
#include <math.h>

typedef __attribute__((ext_vector_type(2))) float v2f;
typedef __attribute__((ext_vector_type(8))) float v8f;

#define NEG_SLOPE 0.2f

static inline int ceil_div(long long a, long long b) { return (int)((a + b - 1) / b); }

// ---------- float atomic max via int ordering trick ----------
__device__ __forceinline__ void atomicMaxF(float* addr, float val) {
  if (val >= 0.0f) {
    atomicMax((int*)addr, __float_as_int(val));
  } else {
    atomicMin((unsigned int*)addr, (unsigned int)__float_as_int(val));
  }
}

// ---------- fill ----------
__global__ void fill_kernel(float* __restrict__ p, float v, int n) {
  int i = blockIdx.x * blockDim.x + threadIdx.x;
  if (i < n) p[i] = v;
}

// ---------- WMMA f32 GEMM: C[M x Nc] = A[M x K] * B[K x Nc], row-major ----------
// One wave per 16x16 tile. Main K-loop is branch-free: row index clamped (garbage
// rows never stored), K remainder handled in a single guarded tail step.
__global__ void gemm_wmma_f32(const float* __restrict__ A, const float* __restrict__ B,
                              float* __restrict__ C, int M, int K, int Nc) {
  const int wavesPerBlock = blockDim.x >> 5;
  const int waveId = blockIdx.x * wavesPerBlock + (threadIdx.x >> 5);
  const int lane   = threadIdx.x & 31;
  const int tilesN = Nc >> 4;
  const int mTiles = (M + 15) >> 4;
  if (waveId >= mTiles * tilesN) return;

  const int m0 = (waveId / tilesN) << 4;
  const int n0 = (waveId % tilesN) << 4;
  const int llo   = lane & 15;
  const int khalf = (lane >> 4) << 1;      // 0 for lanes 0-15, 2 for lanes 16-31
  int row = m0 + llo;
  if (row > M - 1) row = M - 1;            // clamp: OOB rows compute garbage, never stored
  const int col = n0 + llo;

  const float* __restrict__ Arow = A + (size_t)row * K;
  v8f c = {};
  int k = 0;
  for (; k + 4 <= K; k += 4) {             // guard-free main loop
    const int ka = k + khalf;
    v2f a, b;
    a.x = Arow[ka];
    a.y = Arow[ka + 1];
    b.x = B[ka * Nc + col];
    b.y = B[(ka + 1) * Nc + col];
    c = __builtin_amdgcn_wmma_f32_16x16x4_f32(false, a, false, b, (short)0, c, false, false);
  }
  if (k < K) {                             // single tail step (K % 4 != 0)
    const int ka = k + khalf;
    v2f a, b;
    a.x = (ka     < K) ? Arow[ka]     : 0.0f;
    a.y = (ka + 1 < K) ? Arow[ka + 1] : 0.0f;
    b.x = (ka     < K) ? B[ka * Nc + col]       : 0.0f;
    b.y = (ka + 1 < K) ? B[(ka + 1) * Nc + col] : 0.0f;
    c = __builtin_amdgcn_wmma_f32_16x16x4_f32(false, a, false, b, (short)0, c, false, false);
  }

  // C/D layout: VGPR i -> (M = m0+i, N = n0+lane&15), +8 rows for lanes 16-31
  const int ro = m0 + ((lane >> 4) << 3);
#pragma unroll
  for (int i = 0; i < 8; ++i) {
    const int r = ro + i;
    if (r < M) C[r * Nc + col] = c[i];
  }
}

// ---------- per-node attention coefficients ----------
__global__ void attn_coef_kernel(const float* __restrict__ h, const float* __restrict__ a_src,
                                 const float* __restrict__ a_dst, float* __restrict__ als,
                                 float* __restrict__ ald, int n, int C) {
  int i = blockIdx.x * blockDim.x + threadIdx.x;
  if (i >= n) return;
  float s = 0.0f, d = 0.0f;
  for (int c = 0; c < C; ++c) {
    float v = h[i * C + c];
    s += v * a_src[c];
    d += v * a_dst[c];
  }
  als[i] = s;
  ald[i] = d;
}

// ---------- layer 3 fused transform (16 -> 2) + coefficients ----------
__global__ void l3_transform_kernel(const float* __restrict__ x, const float* __restrict__ W,
                                    const float* __restrict__ a_src, const float* __restrict__ a_dst,
                                    float* __restrict__ h, float* __restrict__ als,
                                    float* __restrict__ ald, int n) {
  int i = blockIdx.x * blockDim.x + threadIdx.x;
  if (i >= n) return;
  float h0 = 0.0f, h1 = 0.0f;
  for (int k = 0; k < 16; ++k) {
    float v = x[i * 16 + k];
    h0 += v * W[k * 2 + 0];
    h1 += v * W[k * 2 + 1];
  }
  h[i * 2 + 0] = h0;
  h[i * 2 + 1] = h1;
  als[i] = h0 * a_src[0] + h1 * a_src[1];
  ald[i] = h0 * a_dst[0] + h1 * a_dst[1];
}

__device__ __forceinline__ float leaky(float v) { return v > 0.0f ? v : NEG_SLOPE * v; }

// ---------- edge pass 1: segment max (optionally caches ev[e]) ----------
__global__ void edge_max_kernel(const int* __restrict__ src, const int* __restrict__ dst,
                                int E, int Etot, const float* __restrict__ als,
                                const float* __restrict__ ald, float* __restrict__ emax,
                                float* __restrict__ ebuf) {
  int e = blockIdx.x * blockDim.x + threadIdx.x;
  if (e >= Etot) return;
  int s, d;
  if (e < E) { s = src[e]; d = dst[e]; } else { s = e - E; d = s; }
  float ev = leaky(als[s] + ald[d]);
  if (ebuf) ebuf[e] = ev;
  atomicMaxF(&emax[d], ev);
}

// ---------- edge pass 2: denominator (rewrites ebuf[e] = exp(ev - emax[d])) ----------
__global__ void edge_denom_kernel(const int* __restrict__ src, const int* __restrict__ dst,
                                  int E, int Etot, const float* __restrict__ als,
                                  const float* __restrict__ ald, const float* __restrict__ emax,
                                  float* __restrict__ denom, float* __restrict__ ebuf) {
  int e = blockIdx.x * blockDim.x + threadIdx.x;
  if (e >= Etot) return;
  int d = (e < E) ? dst[e] : e - E;
  float ev;
  if (ebuf) {
    ev = ebuf[e];
  } else {
    int s = (e < E) ? src[e] : e - E;
    ev = leaky(als[s] + ald[d]);
  }
  float ee = __expf(ev - emax[d]);
  if (ebuf) ebuf[e] = ee;
  atomicAdd(&denom[d], ee);
}

// ---------- edge pass 3: alpha-weighted scatter; lane = channel for coalescing ----------
__global__ void edge_scatter_kernel(const int* __restrict__ src, const int* __restrict__ dst,
                                    int E, int Etot, const float* __restrict__ als,
                                    const float* __restrict__ ald, const float* __restrict__ emax,
                                    const float* __restrict__ denom, const float* __restrict__ ebuf,
                                    const float* __restrict__ h, float* __restrict__ agg, int C) {
  int tid = blockIdx.x * blockDim.x + threadIdx.x;
  if (tid >= Etot * C) return;
  int e = tid / C;
  int c = tid - e * C;
  int s, d;
  if (e < E) { s = src[e]; d = dst[e]; } else { s = e - E; d = s; }
  float alpha;
  if (ebuf) {
    alpha = ebuf[e] / denom[d];
  } else {
    float ev = leaky(als[s] + ald[d]);
    alpha = __expf(ev - emax[d]) / denom[d];
  }
  atomicAdd(&agg[d * C + c], h[s * C + c] * alpha);
}

// ---------- bias + relu (in place) ----------
__global__ void bias_relu_kernel(float* __restrict__ agg, const float* __restrict__ b, int n, int C) {
  int i = blockIdx.x * blockDim.x + threadIdx.x;
  if (i >= n * C) return;
  int c = i % C;
  float v = agg[i] + b[c];
  agg[i] = v > 0.0f ? v : 0.0f;
}

// ---------- final: bias + log_softmax over 2 classes ----------
__global__ void final_kernel(const float* __restrict__ agg, const float* __restrict__ b,
                             float* __restrict__ out, int n) {
  int i = blockIdx.x * blockDim.x + threadIdx.x;
  if (i >= n) return;
  float v0 = agg[i * 2 + 0] + b[0];
  float v1 = agg[i * 2 + 1] + b[1];
  float m  = fmaxf(v0, v1);
  float lse = m + __logf(__expf(v0 - m) + __expf(v1 - m));
  out[i * 2 + 0] = v0 - lse;
  out[i * 2 + 1] = v1 - lse;
}

extern "C" void kernel_launch(void* const* d_in, const int* in_sizes, int n_in,
                              void* d_out, int out_size, void* d_ws, size_t ws_size,
                              hipStream_t stream) {
  const float* x   = (const float*)d_in[0];
  const int*   ei  = (const int*)  d_in[1];
  const float* W1  = (const float*)d_in[2];
  const float* as1 = (const float*)d_in[3];
  const float* ad1 = (const float*)d_in[4];
  const float* b1  = (const float*)d_in[5];
  const float* W2  = (const float*)d_in[6];
  const float* as2 = (const float*)d_in[7];
  const float* ad2 = (const float*)d_in[8];
  const float* b2  = (const float*)d_in[9];
  const float* W3  = (const float*)d_in[10];
  const float* as3 = (const float*)d_in[11];
  const float* ad3 = (const float*)d_in[12];
  const float* b3  = (const float*)d_in[13];

  const int N    = in_sizes[0] / 55;
  const int E    = in_sizes[1] / 2;
  const int Etot = E + N;
  const int* src = ei;
  const int* dst = ei + E;

  // workspace layout (floats)
  float* ws    = (float*)d_ws;
  float* h1    = ws;                       // 32N
  float* agg1  = h1   + (size_t)32 * N;    // 32N  (becomes layer-2 input)
  float* h2    = agg1 + (size_t)32 * N;    // 16N
  float* agg2  = h2   + (size_t)16 * N;    // 16N  (becomes layer-3 input)
  float* h3    = agg2 + (size_t)16 * N;    // 2N
  float* agg3  = h3   + (size_t)2  * N;    // 2N
  float* als   = agg3 + (size_t)2  * N;    // N  (reused per layer)
  float* ald   = als  + N;                 // N
  float* emax  = ald  + N;                 // N
  float* denom = emax + N;                 // N

  // optional edge cache (Etot floats) if workspace is big enough
  size_t baseFloats = (size_t)104 * N;
  float* ebuf = (ws_size >= (baseFloats + (size_t)Etot) * sizeof(float))
                    ? (denom + N) : (float*)nullptr;

  const int T = 256;
  const int wavesPerBlock = T / 32;

  // ================= Layer 1: 55 -> 32 =================
  {
    const int C = 32;
    int tiles = ((N + 15) / 16) * (C / 16);
    gemm_wmma_f32<<<ceil_div(tiles, wavesPerBlock), T, 0, stream>>>(x, W1, h1, N, 55, C);
    attn_coef_kernel<<<ceil_div(N, T), T, 0, stream>>>(h1, as1, ad1, als, ald, N, C);
    fill_kernel<<<ceil_div(N, T), T, 0, stream>>>(emax, -INFINITY, N);
    fill_kernel<<<ceil_div(N, T), T, 0, stream>>>(denom, 0.0f, N);
    fill_kernel<<<ceil_div((long long)N * C, T), T, 0, stream>>>(agg1, 0.0f, N * C);
    edge_max_kernel  <<<ceil_div(Etot, T), T, 0, stream>>>(src, dst, E, Etot, als, ald, emax, ebuf);
    edge_denom_kernel<<<ceil_div(Etot, T), T, 0, stream>>>(src, dst, E, Etot, als, ald, emax, denom, ebuf);
    edge_scatter_kernel<<<ceil_div((long long)Etot * C, T), T, 0, stream>>>(
        src, dst, E, Etot, als, ald, emax, denom, ebuf, h1, agg1, C);
    bias_relu_kernel<<<ceil_div((long long)N * C, T), T, 0, stream>>>(agg1, b1, N, C);
  }

  // ================= Layer 2: 32 -> 16 =================
  {
    const int C = 16;
    int tiles = ((N + 15) / 16) * (C / 16);
    gemm_wmma_f32<<<ceil_div(tiles, wavesPerBlock), T, 0, stream>>>(agg1, W2, h2, N, 32, C);
    attn_coef_kernel<<<ceil_div(N, T), T, 0, stream>>>(h2, as2, ad2, als, ald, N, C);
    fill_kernel<<<ceil_div(N, T), T, 0, stream>>>(emax, -INFINITY, N);
    fill_kernel<<<ceil_div(N, T), T, 0, stream>>>(denom, 0.0f, N);
    fill_kernel<<<ceil_div((long long)N * C, T), T, 0, stream>>>(agg2, 0.0f, N * C);
    edge_max_kernel  <<<ceil_div(Etot, T), T, 0, stream>>>(src, dst, E, Etot, als, ald, emax, ebuf);
    edge_denom_kernel<<<ceil_div(Etot, T), T, 0, stream>>>(src, dst, E, Etot, als, ald, emax, denom, ebuf);
    edge_scatter_kernel<<<ceil_div((long long)Etot * C, T), T, 0, stream>>>(
        src, dst, E, Etot, als, ald, emax, denom, ebuf, h2, agg2, C);
    bias_relu_kernel<<<ceil_div((long long)N * C, T), T, 0, stream>>>(agg2, b2, N, C);
  }

  // ================= Layer 3: 16 -> 2, then log_softmax =================
  {
    const int C = 2;
    l3_transform_kernel<<<ceil_div(N, T), T, 0, stream>>>(agg2, W3, as3, ad3, h3, als, ald, N);
    fill_kernel<<<ceil_div(N, T), T, 0, stream>>>(emax, -INFINITY, N);
    fill_kernel<<<ceil_div(N, T), T, 0, stream>>>(denom, 0.0f, N);
    fill_kernel<<<ceil_div((long long)N * C, T), T, 0, stream>>>(agg3, 0.0f, N * C);
    edge_max_kernel  <<<ceil_div(Etot, T), T, 0, stream>>>(src, dst, E, Etot, als, ald, emax, ebuf);
    edge_denom_kernel<<<ceil_div(Etot, T), T, 0, stream>>>(src, dst, E, Etot, als, ald, emax, denom, ebuf);
    edge_scatter_kernel<<<ceil_div((long long)Etot * C, T), T, 0, stream>>>(
        src, dst, E, Etot, als, ald, emax, denom, ebuf, h3, agg3, C);
    final_kernel<<<ceil_div(N, T), T, 0, stream>>>(agg3, b3, (float*)d_out, N);
  }
}